// fbp_14035953123379
// MI455X (gfx1250) — compile-verified
//
#include <hip/hip_runtime.h>
#include <hip/hip_bf16.h>
#include <math.h>

typedef __attribute__((ext_vector_type(2))) float v2f;
typedef __attribute__((ext_vector_type(8))) float v8f;

#define DET 256
#define NANG 512
#define NB 2
#define PADN 512
#define PI_F 3.14159265358979323846f

// Spatial-domain ramp filter kernel h[m], m in [0, 512).
// h[0] = 0.5, h[m odd] = -2/(pi*min(m,512-m))^2, h[m even>0] = 0.
__device__ __forceinline__ float ramp_h(int m) {
    if (m == 0) return 0.5f;
    if (m & 1) {
        int mm = (m < PADN - m) ? m : (PADN - m);
        float pm = PI_F * (float)mm;
        return -2.0f / (pm * pm);
    }
    return 0.0f;
}

// 32-bit LDS byte offset of a __shared__ object (for async-to-LDS dst VGPR).
__device__ __forceinline__ unsigned lds_addr_of(const void* p) {
    return (unsigned)(unsigned long long)(__attribute__((address_space(3))) const char*)p;
}

// -----------------------------------------------------------------------------
// Stage 1: ramp filtering as a GEMM using V_WMMA_F32_16X16X4_F32.
//   fs[b][a][d] = sum_{k=0..255} h[(d-k)&511] * x[b][0][k][a]
// One wave computes a 16(d) x 16(a) tile, K=256 in steps of 4.
// Filter values tabulated once into LDS, *index-reversed*:
//   g[i] = h[(768-i) & 511], so the lane's A pair is the ascending adjacent
//   pair (g[i], g[i+1]) with i = 768-(row-k0) stepping +4: a single
//   ds_load_2addr in register order, no swap movs, no divides in the loop.
// -----------------------------------------------------------------------------
__global__ void __launch_bounds__(32)
fbp_filter_wmma(const float* __restrict__ x, float* __restrict__ fs) {
    __shared__ float g[1024];

    const int tm   = blockIdx.x * 16;          // detector-output tile (M)
    const int tn   = blockIdx.y * 16;          // angle tile (N)
    const int b    = blockIdx.z;
    const int lane = threadIdx.x;              // 0..31 (wave32)
    const int half = lane >> 4;                // 0 or 1
    const int l16  = lane & 15;

    for (int i = lane; i < 1024; i += 32)
        g[i] = ramp_h((768 - i) & (PADN - 1));
    __syncthreads();

    const float* X = x + (size_t)b * DET * NANG;   // X[k][a], row stride NANG
    const int row = tm + l16;                      // A-matrix row (M)
    const int col = tn + l16;                      // B-matrix col (N)

    // i = 768 - (row - k0), k0 = k + 2*half; ranges over [513, 1022].
    int i = 768 - row + 2 * half;

    v8f c = {};
#pragma unroll 8
    for (int k = 0; k < DET; k += 4, i += 4) {
        v2f a;
        a.x = g[i];                                // h[(row-k0)   & 511]
        a.y = g[i + 1];                            // h[(row-k0-1) & 511]
        v2f bb;
        bb.x = X[(size_t)(k + 2 * half)     * NANG + col];
        bb.y = X[(size_t)(k + 2 * half + 1) * NANG + col];
        // 8 args: (neg_a, A, neg_b, B, c_mod, C, reuse_a, reuse_b)
        c = __builtin_amdgcn_wmma_f32_16x16x4_f32(
                false, a, false, bb, (short)0, c, false, false);
    }

    // D layout: lane l, vgpr r -> M = r + 8*(l>>4), N = l&15.
    // Store transposed into fs[b][a][d]: 8 contiguous floats per lane.
    float* outcol = fs + ((size_t)(b * NANG + col)) * DET;
#pragma unroll
    for (int r = 0; r < 8; ++r) {
        const int d = tm + r + half * 8;
        outcol[d] = c[r];
    }
}

// -----------------------------------------------------------------------------
// Stage 2: backprojection. Block = 16x16 pixel tile. Per angle, the 256-float
// filtered row is DMA'd straight into LDS with global_load_async_to_lds_b32
// (ASYNCcnt), double-buffered so angle a+1's row transfer overlaps angle a's
// bilinear gather/accumulate. sin/cos tables precomputed once per block.
// -----------------------------------------------------------------------------
__global__ void __launch_bounds__(256)
fbp_backproject(const float* __restrict__ fs, float* __restrict__ out) {
    __shared__ float rowbuf[2][DET];
    __shared__ float scos[NANG];
    __shared__ float ssin[NANG];

    const int b   = blockIdx.z;
    const int px  = blockIdx.x * 16 + threadIdx.x;
    const int py  = blockIdx.y * 16 + threadIdx.y;
    const int tid = threadIdx.y * 16 + threadIdx.x;    // 0..255

    // Precompute angle tables (2 angles per thread).
    for (int i = tid; i < NANG; i += 256) {
        float s, cth;
        sincosf((float)i * (PI_F / (float)NANG), &s, &cth);
        ssin[i] = s;
        scos[i] = cth;
    }

    const float gx = -1.0f + 2.0f * (float)px / (float)(DET - 1);
    const float gy = -1.0f + 2.0f * (float)py / (float)(DET - 1);
    const float* S = fs + (size_t)b * NANG * DET;      // uniform SGPR base

    const unsigned lds0 = lds_addr_of(&rowbuf[0][tid]);
    const unsigned lds1 = lds_addr_of(&rowbuf[1][tid]);

    // Issue row 0 -> rowbuf[0].
    {
        unsigned voff = (unsigned)(tid * 4);
        asm volatile("global_load_async_to_lds_b32 %0, %1, %2"
                     :: "v"(lds0), "v"(voff), "s"(S) : "memory");
    }

    float acc = 0.0f;
    for (int a = 0; a < NANG; ++a) {
        // Wait for this wave's outstanding async row, then make it visible
        // to the whole workgroup.
        asm volatile("s_wait_asynccnt 0x0" ::: "memory");
        __syncthreads();

        // Prefetch next row into the other buffer. Safe: that buffer was
        // last read in iteration a-1, which completed before this barrier.
        if (a + 1 < NANG) {
            unsigned dst  = ((a + 1) & 1) ? lds1 : lds0;
            unsigned voff = (unsigned)((a + 1) * DET + tid) * 4u;
            asm volatile("global_load_async_to_lds_b32 %0, %1, %2"
                         :: "v"(dst), "v"(voff), "s"(S) : "memory");
        }

        const float* row = rowbuf[a & 1];
        const float t  = gx * scos[a] - gy * ssin[a];
        const float iy = (t + 1.0f) * 0.5f * (float)(DET - 1);
        const float fl = floorf(iy);
        const float w1 = iy - fl;
        const float w0 = 1.0f - w1;
        const int i0 = (int)fl;
        const int i1 = i0 + 1;
        float v = 0.0f;
        if ((unsigned)i0 < (unsigned)DET) v += w0 * row[i0];
        if ((unsigned)i1 < (unsigned)DET) v += w1 * row[i1];
        acc += v;
    }

    out[(size_t)b * DET * DET + (size_t)py * DET + px] =
        acc * (PI_F / (2.0f * (float)NANG));
}

extern "C" void kernel_launch(void* const* d_in, const int* in_sizes, int n_in,
                              void* d_out, int out_size, void* d_ws, size_t ws_size,
                              hipStream_t stream) {
    (void)in_sizes; (void)n_in; (void)out_size; (void)ws_size;
    const float* x  = (const float*)d_in[0];   // (B,1,DET,NANG) fp32 sinogram
    float*       fs = (float*)d_ws;            // (B,NANG,DET) filtered sinogram, 1 MB
    float*       y  = (float*)d_out;           // (B,1,DET,DET)

    dim3 g1(DET / 16, NANG / 16, NB);          // 16 x 32 x 2 tiles
    fbp_filter_wmma<<<g1, 32, 0, stream>>>(x, fs);

    dim3 g2(DET / 16, DET / 16, NB);
    dim3 b2(16, 16);
    fbp_backproject<<<g2, b2, 0, stream>>>(fs, y);
}